// GCNLayer_51419348468394
// MI455X (gfx1250) — compile-verified
//
#include <hip/hip_runtime.h>
#include <hip/hip_bf16.h>

#define N_NODES 50000
#define N_EDGES 800000
#define D_FEAT  64

#define CHUNK   512      // edges staged in LDS per block (multiple of 4)
#define BLOCK   256      // 8 wave32s

// ---- CDNA5 async global->LDS copy engine (ASYNCcnt) ----------------------
#if defined(__has_builtin)
#if __has_builtin(__builtin_amdgcn_global_load_async_to_lds_b32)
#define HAS_ASYNC_LDS 1
#endif
#if __has_builtin(__builtin_amdgcn_global_load_async_to_lds_b128)
#define HAS_ASYNC_LDS_B128 1
#endif
#endif

typedef int v4i_t __attribute__((vector_size(16)));

typedef __attribute__((address_space(1))) int   glb_i32_t;   // global i32
typedef __attribute__((address_space(3))) int   lds_i32_t;   // LDS i32
typedef __attribute__((address_space(1))) v4i_t glb_v4i_t;   // global v4i
typedef __attribute__((address_space(3))) v4i_t lds_v4i_t;   // LDS v4i

__device__ __forceinline__ void async_copy_b32(const void* gsrc, void* ldst) {
#ifdef HAS_ASYNC_LDS
    __builtin_amdgcn_global_load_async_to_lds_b32(
        (glb_i32_t*)gsrc, (lds_i32_t*)ldst, /*offset=*/0, /*cpol=*/0);
#else
    *(int*)ldst = *(const int*)gsrc;   // fallback: global_load + ds_store
#endif
}

// 16-byte async copy; caller guarantees 16B alignment of both sides.
__device__ __forceinline__ void async_copy_b128(const void* gsrc, void* ldst) {
#ifdef HAS_ASYNC_LDS_B128
    __builtin_amdgcn_global_load_async_to_lds_b128(
        (glb_v4i_t*)gsrc, (lds_v4i_t*)ldst, /*offset=*/0, /*cpol=*/0);
#else
    async_copy_b32((const int*)gsrc + 0, (int*)ldst + 0);
    async_copy_b32((const int*)gsrc + 1, (int*)ldst + 1);
    async_copy_b32((const int*)gsrc + 2, (int*)ldst + 2);
    async_copy_b32((const int*)gsrc + 3, (int*)ldst + 3);
#endif
}

__device__ __forceinline__ void wait_async_copies() {
#if defined(HAS_ASYNC_LDS) || defined(HAS_ASYNC_LDS_B128)
#if __has_builtin(__builtin_amdgcn_s_wait_asynccnt)
    __builtin_amdgcn_s_wait_asynccnt(0);
#else
    asm volatile("s_wait_asynccnt 0" ::: "memory");
#endif
#endif
}

// ---- SpMM: out[row[e]] += val[e] * embeds[col[e]]  (first => in disasm) ---

__global__ __launch_bounds__(BLOCK)
void gcn_spmm_kernel(const int*   __restrict__ erow,
                     const int*   __restrict__ ecol,
                     const float* __restrict__ eval,
                     const float* __restrict__ emb,
                     float*       __restrict__ out) {
    __shared__ alignas(16) int   s_row[CHUNK];
    __shared__ alignas(16) int   s_col[CHUNK];
    __shared__ alignas(16) float s_val[CHUNK];

    const int tid  = threadIdx.x;
    const int base = blockIdx.x * CHUNK;
    const int rem  = N_EDGES - base;
    const int n    = rem < CHUNK ? rem : CHUNK;   // always a multiple of 4

    // Stage this block's edge triples into LDS with 16B async copies.
    for (int i = tid * 4; i < n; i += BLOCK * 4) {
        async_copy_b128(&erow[base + i], &s_row[i]);
        async_copy_b128(&ecol[base + i], &s_col[i]);
        async_copy_b128(&eval[base + i], &s_val[i]);
    }
    wait_async_copies();
    __syncthreads();

    // One wave32 per edge; each lane owns a float2 of the 64 features.
    const int sub = tid & 31;   // lane -> feature pair
    const int grp = tid >> 5;   // wave index in block (0..7)
    const float2* __restrict__ emb2 = (const float2*)emb;

#pragma unroll 2
    for (int e = grp; e < n; e += BLOCK / 32) {
        const int   r = s_row[e];            // broadcast LDS reads
        const int   c = s_col[e];
        const float v = s_val[e];
        // coalesced 256B gather of one embedding row per wave (L2 resident)
        const float2 g = emb2[c * (D_FEAT / 2) + sub];
        float* o = out + r * D_FEAT + sub * 2;
        atomicAdd(o,     v * g.x);           // contiguous-lane f32 atomics
        atomicAdd(o + 1, v * g.y);
    }
}

__global__ void gcn_zero_kernel(float4* __restrict__ out, int n4) {
    int i = blockIdx.x * blockDim.x + threadIdx.x;
    if (i < n4) out[i] = float4{0.f, 0.f, 0.f, 0.f};
}

extern "C" void kernel_launch(void* const* d_in, const int* in_sizes, int n_in,
                              void* d_out, int out_size, void* d_ws, size_t ws_size,
                              hipStream_t stream) {
    (void)in_sizes; (void)n_in; (void)d_ws; (void)ws_size;

    const int*   erow = (const int*)  d_in[0];
    const int*   ecol = (const int*)  d_in[1];
    const float* eval = (const float*)d_in[2];
    const float* emb  = (const float*)d_in[3];
    float*       out  = (float*)d_out;

    // 1) zero the accumulator (atomic-accumulated output; harness poisons it)
    const int n4 = out_size / 4;                       // 3.2M floats -> 800K float4
    gcn_zero_kernel<<<(n4 + BLOCK - 1) / BLOCK, BLOCK, 0, stream>>>((float4*)out, n4);

    // 2) edge-parallel SpMM with async-LDS-staged edges + f32 scatter-add
    const int nblocks = (N_EDGES + CHUNK - 1) / CHUNK; // 1563
    gcn_spmm_kernel<<<nblocks, BLOCK, 0, stream>>>(erow, ecol, eval, emb, out);
}